// RECT_L_50594714747240
// MI455X (gfx1250) — compile-verified
//
#include <hip/hip_runtime.h>
#include <hip/hip_bf16.h>
#include <math.h>

#define N_NODES 100000
#define N_EDGES 1600000
#define IN_DIM  256
#define HID     128

typedef __attribute__((ext_vector_type(2))) float v2f;
typedef __attribute__((ext_vector_type(8))) float v8f;

// ---------------------------------------------------------------------------
// CSR construction: counts -> exclusive scan -> scatter {src, norm} by dst
// ---------------------------------------------------------------------------
__global__ void counts_init_kernel(int* __restrict__ counts, int n) {
    int i = blockIdx.x * blockDim.x + threadIdx.x;
    if (i < n) counts[i] = 0;
}

__global__ void counts_kernel(const int* __restrict__ dst, int* __restrict__ counts, int e) {
    int i = blockIdx.x * blockDim.x + threadIdx.x;
    if (i < e) atomicAdd(&counts[dst[i]], 1);
}

// Single-workgroup chunked Hillis-Steele exclusive scan (100k elems, us-scale;
// edge phases dominate by orders of magnitude).
__global__ void scan_kernel(const int* __restrict__ counts,
                            int* __restrict__ row_ptr,
                            int* __restrict__ cursor, int n) {
    __shared__ int lds[1024];
    __shared__ int carry_s;
    const int tid = threadIdx.x;
    if (tid == 0) carry_s = 0;
    __syncthreads();
    for (int base = 0; base < n; base += 1024) {
        const int i = base + tid;
        const int v = (i < n) ? counts[i] : 0;
        lds[tid] = v;
        __syncthreads();
        for (int off = 1; off < 1024; off <<= 1) {
            int t = (tid >= off) ? lds[tid - off] : 0;
            __syncthreads();
            lds[tid] += t;
            __syncthreads();
        }
        const int incl  = lds[tid];
        const int carry = carry_s;
        if (i < n) {
            const int excl = carry + incl - v;
            row_ptr[i] = excl;
            cursor[i]  = excl;
        }
        __syncthreads();
        if (tid == 1023) carry_s = carry + incl;
        __syncthreads();
    }
    if (tid == 0) row_ptr[n] = carry_s;
}

__global__ void dinv_kernel(const int* __restrict__ counts, float* __restrict__ dinv, int n) {
    int i = blockIdx.x * blockDim.x + threadIdx.x;
    if (i < n) dinv[i] = rsqrtf((float)counts[i] + 1.0f);   // +1 = self-loop
}

// Scatter interleaved {src, norm} records: the gather loop then needs exactly
// one b64 load per edge instead of two dependent dword loads.
__global__ void scatter_kernel(const int* __restrict__ src, const int* __restrict__ dst,
                               const float* __restrict__ dinv,
                               int* __restrict__ cursor, int2* __restrict__ edata, int e) {
    int i = blockIdx.x * blockDim.x + threadIdx.x;
    if (i < e) {
        const int s = src[i];
        const int d = dst[i];
        const int pos = atomicAdd(&cursor[d], 1);
        const float nrm = dinv[s] * dinv[d];
        int2 rec;
        rec.x = s;
        rec.y = __float_as_int(nrm);
        edata[pos] = rec;
    }
}

// ---------------------------------------------------------------------------
// GEMM1: h[N,128] = x[N,256] @ W_conv[256,128]   (f32 WMMA 16x16x4)
// ---------------------------------------------------------------------------
__global__ void gemm1_kernel(const float* __restrict__ x,
                             const float* __restrict__ W,
                             float* __restrict__ h) {
    const int wave = threadIdx.x >> 5;
    const int lane = threadIdx.x & 31;
    const int half = lane >> 4;
    const int l16  = lane & 15;

    const int m0 = blockIdx.x * 16;    // 6250 row tiles exactly
    const int n0 = wave * 16;          // 8 waves * 16 = 128

    const int arow = m0 + l16;
    const int bcol = n0 + l16;

    v8f acc = {};
    for (int k = 0; k < IN_DIM; k += 4) {
        const int ka = k + half * 2;
        const float* ap = x + (size_t)arow * IN_DIM + ka;
        v2f a; a.x = ap[0]; a.y = ap[1];
        v2f b;
        b.x = W[(size_t)ka * HID + bcol];
        b.y = W[(size_t)(ka + 1) * HID + bcol];
        acc = __builtin_amdgcn_wmma_f32_16x16x4_f32(false, a, false, b,
                                                    (short)0, acc, false, false);
    }
#pragma unroll
    for (int r = 0; r < 8; ++r) {
        const int row = m0 + r + half * 8;
        h[(size_t)row * HID + n0 + l16] = acc[r];
    }
}

// ---------------------------------------------------------------------------
// Aggregation (atomic-free gather): one wave per node, lane owns 4 columns.
// agg[n] = dinv[n]^2 * h[n] + sum_{(s,w) in CSR(n)} w * h[s]
// h (51.2 MB) is L2-resident; software-pipelined edge records + prefetch of
// the next h row overlap L2 latency with the FMAs.
// ---------------------------------------------------------------------------
__global__ void agg_kernel(const int* __restrict__ row_ptr,
                           const int2* __restrict__ edata,
                           const float* __restrict__ h,
                           const float* __restrict__ dinv,
                           float* __restrict__ agg) {
    long long gid = (long long)blockIdx.x * blockDim.x + threadIdx.x;
    const int node = (int)(gid >> 5);
    const int lane = (int)(gid & 31);
    if (node >= N_NODES) return;

    const float dn = dinv[node];
    const int c0 = lane * 4;

    const float4 hn = *(const float4*)(h + (size_t)node * HID + c0);
    const float w0 = dn * dn;                      // self-loop
    float4 acc;
    acc.x = hn.x * w0; acc.y = hn.y * w0; acc.z = hn.z * w0; acc.w = hn.w * w0;

    const int beg = row_ptr[node];
    const int end = row_ptr[node + 1];

    int2 nxt;
    if (beg < end) nxt = edata[beg];
    for (int j = beg; j < end; ++j) {
        const int2 cur = nxt;
        if (j + 1 < end) {
            nxt = edata[j + 1];
            // prefetch the next neighbor's h row (global_prefetch_b8)
            __builtin_prefetch(h + (size_t)nxt.x * HID + c0, 0, 3);
        }
        const int   s = cur.x;
        const float w = __int_as_float(cur.y);
        const float4 hv = *(const float4*)(h + (size_t)s * HID + c0);
        acc.x = fmaf(hv.x, w, acc.x);
        acc.y = fmaf(hv.y, w, acc.y);
        acc.z = fmaf(hv.z, w, acc.z);
        acc.w = fmaf(hv.w, w, acc.w);
    }
    *(float4*)(agg + (size_t)node * HID + c0) = acc;
}

// ---------------------------------------------------------------------------
// GEMM2: out[N,256] = PReLU(agg + b_conv) @ W_lin[128,256] + b_lin
// ---------------------------------------------------------------------------
__global__ void gemm2_kernel(const float* __restrict__ agg,
                             const float* __restrict__ bconv,
                             const float* __restrict__ prelu_a,
                             const float* __restrict__ Wlin,
                             const float* __restrict__ blin,
                             float* __restrict__ out) {
    const int wave = threadIdx.x >> 5;
    const int lane = threadIdx.x & 31;
    const int half = lane >> 4;
    const int l16  = lane & 15;

    const int m0 = blockIdx.x * 16;
    const int n0 = (blockIdx.y * 8 + wave) * 16;

    const float slope = prelu_a[0];
    const int arow = m0 + l16;
    const int bcol = n0 + l16;

    v8f acc = {};
    for (int k = 0; k < HID; k += 4) {
        const int ka = k + half * 2;
        float a0 = agg[(size_t)arow * HID + ka]     + bconv[ka];
        float a1 = agg[(size_t)arow * HID + ka + 1] + bconv[ka + 1];
        a0 = (a0 >= 0.0f) ? a0 : slope * a0;   // PReLU
        a1 = (a1 >= 0.0f) ? a1 : slope * a1;
        v2f a; a.x = a0; a.y = a1;
        v2f b;
        b.x = Wlin[(size_t)ka * IN_DIM + bcol];
        b.y = Wlin[(size_t)(ka + 1) * IN_DIM + bcol];
        acc = __builtin_amdgcn_wmma_f32_16x16x4_f32(false, a, false, b,
                                                    (short)0, acc, false, false);
    }
#pragma unroll
    for (int r = 0; r < 8; ++r) {
        const int row = m0 + r + half * 8;
        const int col = n0 + l16;
        out[(size_t)row * IN_DIM + col] = acc[r] + blin[col];
    }
}

// ---------------------------------------------------------------------------
extern "C" void kernel_launch(void* const* d_in, const int* in_sizes, int n_in,
                              void* d_out, int out_size, void* d_ws, size_t ws_size,
                              hipStream_t stream) {
    const float* x     = (const float*)d_in[0];           // [N, 256]
    const int*   ei    = (const int*)d_in[1];             // [2, E]
    const float* Wconv = (const float*)d_in[2];           // [256, 128]
    const float* bconv = (const float*)d_in[3];           // [128]
    const float* pa    = (const float*)d_in[4];           // [1]
    const float* Wlin  = (const float*)d_in[5];           // [128, 256]
    const float* blin  = (const float*)d_in[6];           // [256]
    float* out = (float*)d_out;                           // [N, 256]

    const int* src = ei;               // row 0
    const int* dst = ei + N_EDGES;     // row 1

    // workspace layout (all 16B-aligned offsets)
    char* ws = (char*)d_ws;
    size_t off = 0;
    float* dinv    = (float*)(ws + off); off += (size_t)N_NODES * 4;            // 400 KB
    float* h       = (float*)(ws + off); off += (size_t)N_NODES * HID * 4;      // 51.2 MB
    float* agg     = (float*)(ws + off); off += (size_t)N_NODES * HID * 4;      // 51.2 MB
    int*   counts  = (int*)  (ws + off); off += (size_t)N_NODES * 4;
    int*   row_ptr = (int*)  (ws + off); off += ((size_t)N_NODES + 4) * 4;      // N+1, padded
    int*   cursor  = (int*)  (ws + off); off += (size_t)N_NODES * 4;
    int2*  edata   = (int2*) (ws + off); off += (size_t)N_EDGES * 8;            // 12.8 MB

    const int T = 256;
    const int gN = (N_NODES + T - 1) / T;
    const int gE = (N_EDGES + T - 1) / T;

    // 1) CSR build
    counts_init_kernel<<<gN, T, 0, stream>>>(counts, N_NODES);
    counts_kernel<<<gE, T, 0, stream>>>(dst, counts, N_EDGES);
    scan_kernel<<<1, 1024, 0, stream>>>(counts, row_ptr, cursor, N_NODES);
    dinv_kernel<<<gN, T, 0, stream>>>(counts, dinv, N_NODES);
    scatter_kernel<<<gE, T, 0, stream>>>(src, dst, dinv, cursor, edata, N_EDGES);

    // 2) h = x @ W_conv  (WMMA f32)
    gemm1_kernel<<<N_NODES / 16, T, 0, stream>>>(x, Wconv, h);

    // 3) atomic-free gather aggregation (wave per node)
    {
        long long tot = (long long)N_NODES * 32;
        agg_kernel<<<(int)((tot + T - 1) / T), T, 0, stream>>>(row_ptr, edata, h, dinv, agg);
    }

    // 4) out = PReLU(agg + b_conv) @ W_lin + b_lin  (WMMA f32)
    gemm2_kernel<<<dim3(N_NODES / 16, 2), T, 0, stream>>>(agg, bconv, pa, Wlin, blin, out);
}